// MinimumSpanning3DTree_28810640622323
// MI455X (gfx1250) — compile-verified
//
#include <hip/hip_runtime.h>
#include <hip/hip_bf16.h>

// Problem constants (from reference)
#define BB    4
#define CC    128
#define HH    256
#define WW    256
#define MIDW  128
#define HWP   (HH * WW)      // 65536, also V
#define VV    65536
#define EE    163072
#define EPSW  1e-8f

// Edge-group offsets (reference concatenation order)
//  G0 L-vert : [0,      32640)   u=i*W+j,       v=u+W   (i<255, j<128)
//  G1 L-horz : [32640,  65152)   u=i*W+j,       v=u+1   (i<256, j<127)
//  G2 R-vert : [65152,  97792)   u=i*W+128+j,   v=u+W
//  G3 R-horz : [97792, 130304)   u=i*W+128+j,   v=u+1
//  G4 cross  : [130304,163072)   u=i*W+j,       v=u+128
__device__ __forceinline__ void edge_uv(int e, int& u, int& v) {
  if (e < 32640)        { int i = e >> 7, j = e & 127;                u = i*WW + j;        v = u + WW; }
  else if (e < 65152)   { int t = e - 32640;  int i = t/127, j = t - i*127; u = i*WW + j;        v = u + 1;  }
  else if (e < 97792)   { int t = e - 65152;  int i = t >> 7, j = t & 127;  u = i*WW + MIDW + j; v = u + WW; }
  else if (e < 130304)  { int t = e - 97792;  int i = t/127, j = t - i*127; u = i*WW + MIDW + j; v = u + 1;  }
  else                  { int t = e - 130304; int i = t >> 7, j = t & 127;  u = i*WW + j;        v = u + MIDW; }
}

typedef float v2f __attribute__((ext_vector_type(2)));
typedef float v8f __attribute__((ext_vector_type(8)));

// ---------------------------------------------------------------------------
// Kernel 1: per-pixel L2 norm over channels.  One HBM pass over guide (128MB).
// ---------------------------------------------------------------------------
__global__ void norms_kernel(const float* __restrict__ g, float* __restrict__ norms) {
  int pix = blockIdx.x * blockDim.x + threadIdx.x;   // 0..V-1 (coalesced)
  int b   = blockIdx.y;
  const float* gb = g + (size_t)b * CC * HWP + pix;
  float acc = 0.f;
#pragma unroll 8
  for (int c = 0; c < CC; ++c) {
    float x = gb[(size_t)c * HWP];
    acc = fmaf(x, x, acc);
  }
  norms[(size_t)b * VV + pix] = sqrtf(acc);
}

// ---------------------------------------------------------------------------
// Kernel 2: edge weights via V_WMMA_F32_16X16X4_F32.
// Each wave handles 16 edges; A rows = u-pixels, B cols = v-pixels; after 32
// K=4 steps the D diagonal holds the 16 f32 dot products.  Weight -> ordered
// 64-bit key (monotone float bits << 32 | edge index) == reference's rank
// ordering with stable tie-break.  Also zero-initializes d_out.
// ---------------------------------------------------------------------------
__global__ void edge_wmma_kernel(const float* __restrict__ g,
                                 const float* __restrict__ norms,
                                 unsigned long long* __restrict__ keys,
                                 float* __restrict__ out) {
  const int lane = threadIdx.x & 31;
  const int wv   = threadIdx.x >> 5;
  const int b    = blockIdx.y;
  const int base = (blockIdx.x * 8 + wv) * 16;     // 8 waves/block, 16 edges/wave
  const int m    = lane & 15;                      // edge slot within wave
  const int e    = base + m;

  int u, v;
  edge_uv(e, u, v);

  const float* gb = g + (size_t)b * CC * HWP;
  const int half2 = (lane >> 4) << 1;              // lanes 0-15 -> K 0,1 ; lanes 16-31 -> K 2,3

  v8f acc = {};
#pragma unroll 4
  for (int kk = 0; kk < CC; kk += 4) {
    const float* pu = gb + (size_t)(kk + half2) * HWP + u;
    const float* pv = gb + (size_t)(kk + half2) * HWP + v;
    v2f av, bv;
    av.x = pu[0]; av.y = pu[HWP];                  // A: M=m in lanes, K in {vgpr, lane-half}
    bv.x = pv[0]; bv.y = pv[HWP];                  // B: N=m in lanes, mirrored K layout
    acc = __builtin_amdgcn_wmma_f32_16x16x4_f32(
        /*neg_a=*/false, av, /*neg_b=*/false, bv,
        /*c_mod=*/(short)0, acc, /*reuse_a=*/false, /*reuse_b=*/false);
  }

  // Diagonal extraction: D(m,m) sits at vgpr m%8, lane m (m<8) or lane m+16 (m>=8).
  int sel = -1;
  bool valid = false;
  if (lane < 8)        { sel = lane;      valid = true; }   // m = lane
  else if (lane >= 24) { sel = lane - 24; valid = true; }   // m = lane - 16
  float dot = acc[0];
#pragma unroll
  for (int gidx = 1; gidx < 8; ++gidx)
    if (sel == gidx) dot = acc[gidx];

  if (valid) {
    float na = norms[(size_t)b * VV + u];
    float nb = norms[(size_t)b * VV + v];
    float w  = dot / fmaxf(na * nb, EPSW);
    unsigned int uw = __float_as_uint(w);
    uw = (uw & 0x80000000u) ? ~uw : (uw | 0x80000000u);    // order-preserving float->uint
    keys[(size_t)b * EE + e] = ((unsigned long long)uw << 32) | (unsigned int)e;
    out [(size_t)b * EE + e] = 0.0f;                       // zero-init selection output
  }
}

// ---------------------------------------------------------------------------
// Kernel 3: 17 Boruvka rounds, one 1024-thread workgroup per batch.
// State (~2.5MB/batch) stays L2-resident.  segment_min == u64 atomicMin on
// component roots; win scatters are unique per root (keys are unique).
// ---------------------------------------------------------------------------
__global__ void __launch_bounds__(1024)
mst_kernel(const unsigned long long* __restrict__ keys_all,
           int* __restrict__ parent_all,
           int* __restrict__ root_all,
           unsigned long long* __restrict__ comp_all,
           float* __restrict__ out) {
  const int b = blockIdx.x;
  const unsigned long long* keys = keys_all + (size_t)b * EE;
  int* parent               = parent_all + (size_t)b * VV;
  int* root                 = root_all   + (size_t)b * VV;
  unsigned long long* comp  = comp_all   + (size_t)b * VV;
  float* sel                = out        + (size_t)b * EE;

  const int t  = threadIdx.x;
  const int NT = blockDim.x;

  for (int i = t; i < VV; i += NT) parent[i] = i;
  __syncthreads();

  for (int r = 0; r < 17; ++r) {
    // 1) full root (reference's 2^17-step pointer doubling fixed point)
    for (int i = t; i < VV; i += NT) {
      int x = i, p = parent[x];
      for (int it = 0; it < 256 && p != x; ++it) { x = p; p = parent[x]; }
      root[i] = x;
    }
    __syncthreads();

    // 2) reset component minima
    for (int i = t; i < VV; i += NT) comp[i] = ~0ull;
    __syncthreads();

    // 3) segment_min over both endpoints
    for (int e = t; e < EE; e += NT) {
      int u, v; edge_uv(e, u, v);
      int ru = root[u], rv = root[v];
      if (ru != rv) {
        unsigned long long k = keys[e];
        atomicMin(&comp[ru], k);
        atomicMin(&comp[rv], k);
      }
    }
    __syncthreads();

    // 4) parent <- root
    for (int i = t; i < VV; i += NT) parent[i] = root[i];
    __syncthreads();

    // 5a) win_u scatter (+ record selections); unique writer per ru
    for (int e = t; e < EE; e += NT) {
      int u, v; edge_uv(e, u, v);
      int ru = root[u], rv = root[v];
      if (ru != rv) {
        unsigned long long k = keys[e];
        bool wu = (comp[ru] == k), wv = (comp[rv] == k);
        if (wu | wv) sel[e] = 1.0f;
        if (wu) parent[ru] = rv;
      }
    }
    __syncthreads();

    // 5b) win_v scatter overrides win_u (reference's p2-after-p1 order)
    for (int e = t; e < EE; e += NT) {
      int u, v; edge_uv(e, u, v);
      int ru = root[u], rv = root[v];
      if (ru != rv) {
        if (comp[rv] == keys[e]) parent[rv] = ru;
      }
    }
    __syncthreads();

    // 6) break 2-cycles: smaller member becomes root (tmp in root[])
    for (int i = t; i < VV; i += NT) {
      int p = parent[i];
      int pp = parent[p];
      root[i] = (pp == i && i < p) ? i : p;
    }
    __syncthreads();
    for (int i = t; i < VV; i += NT) parent[i] = root[i];
    __syncthreads();
  }
}

// ---------------------------------------------------------------------------
extern "C" void kernel_launch(void* const* d_in, const int* in_sizes, int n_in,
                              void* d_out, int out_size, void* d_ws, size_t ws_size,
                              hipStream_t stream) {
  const float* guide = (const float*)d_in[0];        // (B,C,H,W) f32
  float* out = (float*)d_out;                        // (B,E) f32

  // Workspace carve-out (~10 MB total)
  size_t off = 0;
  auto carve = [&](size_t bytes) -> void* {
    void* p = (char*)d_ws + off;
    off += (bytes + 255) & ~(size_t)255;
    return p;
  };
  float*              norms  = (float*)             carve((size_t)BB * VV * sizeof(float));
  unsigned long long* keys   = (unsigned long long*)carve((size_t)BB * EE * sizeof(unsigned long long));
  int*                parent = (int*)               carve((size_t)BB * VV * sizeof(int));
  int*                root   = (int*)               carve((size_t)BB * VV * sizeof(int));
  unsigned long long* comp   = (unsigned long long*)carve((size_t)BB * VV * sizeof(unsigned long long));
  (void)ws_size; (void)in_sizes; (void)n_in; (void)out_size;

  // 1) per-pixel norms: one HBM pass over guide
  norms_kernel<<<dim3(VV / 256, BB), 256, 0, stream>>>(guide, norms);

  // 2) edge dots via f32 WMMA (16 edges / wave, diag of 16x16 product)
  edge_wmma_kernel<<<dim3(EE / 128, BB), 256, 0, stream>>>(guide, norms, keys, out);

  // 3) 17 Boruvka rounds, one workgroup per batch (state L2-resident)
  mst_kernel<<<BB, 1024, 0, stream>>>(keys, parent, root, comp, out);
}